// EchoStateNetwork_53480932770456
// MI455X (gfx1250) — compile-verified
//
#include <hip/hip_runtime.h>
#include <math.h>

// ESN dims
#define ESN_H 4096
#define ESN_I 64
#define ESN_O 64
#define ESN_S 4096
#define ESN_K_EXT (1 + ESN_I + ESN_H)   // 4161
#define SCAN_NB 128                     // persistent blocks for the scan

typedef __attribute__((ext_vector_type(2))) float v2f;
typedef __attribute__((ext_vector_type(8))) float v8f;

// ---------------------------------------------------------------------------
// Kernel 1: proj[s][h] = sum_i inputs[s][i] * W_in[h][i]
// fp32 WMMA 16x16x4. One wave per 16x16 output tile, K = 64 -> 16 WMMAs.
// Writes proj directly into the "states" region of d_out (overwritten in-place
// by the scan kernel with tanh states).
// ---------------------------------------------------------------------------
__global__ __launch_bounds__(128) void esn_proj_wmma(
    const float* __restrict__ inputs,   // (S, I) row-major
    const float* __restrict__ W_in,     // (H, I) row-major
    float* __restrict__ states)         // (S, H) dest for proj
{
  const int lane  = threadIdx.x & 31;
  const int gwave = blockIdx.x * 4 + (threadIdx.x >> 5);  // 65536 waves
  const int s0 = (gwave >> 8) * 16;                       // 256 tiles in S
  const int h0 = (gwave & 255) * 16;                      // 256 tiles in H
  const int mn = lane & 15;
  const int kb = (lane >> 4) << 1;       // K sub-offset: lanes 0-15 -> 0, 16-31 -> 2
  const int mbase = (lane >> 4) * 8;     // D rows: lanes 0-15 -> M=v, 16-31 -> M=v+8

  const float* arow = inputs + (s0 + mn) * ESN_I + kb;   // A[m][k]
  const float* brow = W_in   + (h0 + mn) * ESN_I + kb;   // B[k][n] = W_in[n][k]

  v8f acc = {};
#pragma unroll
  for (int kk = 0; kk < ESN_I; kk += 4) {
    v2f a, b;
    a.x = arow[kk]; a.y = arow[kk + 1];
    b.x = brow[kk]; b.y = brow[kk + 1];
    acc = __builtin_amdgcn_wmma_f32_16x16x4_f32(false, a, false, b,
                                                (short)0, acc, false, false);
  }
#pragma unroll
  for (int v = 0; v < 8; ++v)
    states[(s0 + mbase + v) * ESN_H + h0 + mn] = acc[v];
}

// ---------------------------------------------------------------------------
// Kernel 2: persistent sequential scan.
//   state_t = tanh(proj_t + W_res @ state_{t-1})
// states[t] initially holds proj_t; overwritten in place with the new state.
// Per step: each of 128 blocks owns 32 rows (4 rows / wave); row dot of length
// 4096 split across 32 lanes as float4 loads (fully coalesced, W_res is
// L2-resident at 64 MB), then a 5-step wave32 xor-shuffle reduction.
// Grid-wide barrier via a monotonically increasing counter in d_ws.
// ---------------------------------------------------------------------------
__global__ __launch_bounds__(256) void esn_scan(
    const float* __restrict__ W_res,    // (H, H) row-major
    float* __restrict__ states,         // (S, H): proj in, states out
    unsigned* __restrict__ ctr)         // barrier counter (zeroed by memset)
{
  const int lane = threadIdx.x & 31;
  const int wave = threadIdx.x >> 5;                  // 0..7
  const int rowBase = blockIdx.x * 32 + wave * 4;     // 4 rows per wave

  for (int t = 0; t < ESN_S; ++t) {
    float vals[4];
    const float* x = states + (t - 1) * ESN_H;        // valid when t > 0
#pragma unroll
    for (int r = 0; r < 4; ++r) {
      float acc = 0.0f;
      if (t > 0) {
        const float4* wrow = (const float4*)(W_res + (rowBase + r) * ESN_H) + lane;
        const float4* xv   = (const float4*)x + lane;
#pragma unroll 4
        for (int j = 0; j < 32; ++j) {                // 32 lanes * 32 * 4 = 4096
          float4 w  = wrow[j * 32];
          float4 xx = xv[j * 32];
          acc += w.x * xx.x + w.y * xx.y + w.z * xx.z + w.w * xx.w;
        }
#pragma unroll
        for (int off = 16; off > 0; off >>= 1)
          acc += __shfl_xor(acc, off, 32);
      }
      vals[r] = acc;
    }
    if (lane == 0) {
#pragma unroll
      for (int r = 0; r < 4; ++r) {
        const int row = rowBase + r;
        float pin = states[t * ESN_H + row];          // proj value
        states[t * ESN_H + row] = tanhf(pin + vals[r]);
      }
    }
    // ---- grid-wide barrier ----
    __threadfence();          // release this wave's stores to device scope
    __syncthreads();
    if (threadIdx.x == 0) {
      __hip_atomic_fetch_add(ctr, 1u, __ATOMIC_RELEASE, __HIP_MEMORY_SCOPE_AGENT);
      const unsigned target = (unsigned)SCAN_NB * (unsigned)(t + 1);
      while (__hip_atomic_load(ctr, __ATOMIC_RELAXED, __HIP_MEMORY_SCOPE_AGENT) < target)
        __builtin_amdgcn_s_sleep(2);
    }
    __syncthreads();
    __threadfence();          // acquire: invalidate WGP$ so states[t] is fresh
  }
}

// ---------------------------------------------------------------------------
// Kernel 3: readout. out[s][o] = W_out[o][0] + inputs[s]·W_out[o][1:65]
//                               + states[s]·W_out[o][65:4161]
// Bias + input part (K=65) done as scalar init of the accumulator fragment;
// states part (K=4096) via fp32 WMMA 16x16x4. One wave per 16x16 tile.
// ---------------------------------------------------------------------------
__global__ __launch_bounds__(128) void esn_readout_wmma(
    const float* __restrict__ inputs,   // (S, I)
    const float* __restrict__ W_out,    // (O, 4161)
    const float* __restrict__ states,   // (S, H)
    float* __restrict__ outputs)        // (S, O)
{
  const int lane  = threadIdx.x & 31;
  const int gwave = blockIdx.x * 4 + (threadIdx.x >> 5);  // 1024 waves
  const int s0 = (gwave >> 2) * 16;                       // 256 tiles in S
  const int o0 = (gwave & 3) * 16;                        // 4 tiles in O
  const int mn = lane & 15;
  const int kb = (lane >> 4) << 1;
  const int mbase = (lane >> 4) * 8;

  // acc[v] corresponds to D[M = mbase+v][N = mn]
  v8f acc;
#pragma unroll
  for (int v = 0; v < 8; ++v) {
    const float* wo = W_out + (o0 + mn) * ESN_K_EXT;
    const float* in = inputs + (s0 + mbase + v) * ESN_I;
    float sum = wo[0];
#pragma unroll 8
    for (int i = 0; i < ESN_I; ++i)
      sum += in[i] * wo[1 + i];
    acc[v] = sum;
  }

  const float* arow = states + (s0 + mn) * ESN_H + kb;
  const float* brow = W_out + (o0 + mn) * ESN_K_EXT + (1 + ESN_I) + kb;
#pragma unroll 4
  for (int kk = 0; kk < ESN_H; kk += 4) {
    v2f a, b;
    a.x = arow[kk]; a.y = arow[kk + 1];
    b.x = brow[kk]; b.y = brow[kk + 1];
    acc = __builtin_amdgcn_wmma_f32_16x16x4_f32(false, a, false, b,
                                                (short)0, acc, false, false);
  }
#pragma unroll
  for (int v = 0; v < 8; ++v)
    outputs[(s0 + mbase + v) * ESN_O + o0 + mn] = acc[v];
}

// ---------------------------------------------------------------------------
extern "C" void kernel_launch(void* const* d_in, const int* in_sizes, int n_in,
                              void* d_out, int out_size, void* d_ws, size_t ws_size,
                              hipStream_t stream) {
  const float* inputs = (const float*)d_in[0];  // (1, S, I)
  const float* W_in   = (const float*)d_in[1];  // (H, I)
  const float* W_res  = (const float*)d_in[2];  // (H, H)
  const float* W_out  = (const float*)d_in[3];  // (O, 1+I+H)

  float* outputs = (float*)d_out;                        // (S, O) = 262144 floats
  float* states  = (float*)d_out + ESN_S * ESN_O;        // (S, H)
  unsigned* ctr  = (unsigned*)d_ws;                      // barrier counter

  // Zero the barrier counter (graph-capture-safe memset node).
  hipMemsetAsync(ctr, 0, 256, stream);

  // 1) input projection -> states region (65536 tiles, 4 waves/block)
  esn_proj_wmma<<<dim3(16384), dim3(128), 0, stream>>>(inputs, W_in, states);

  // 2) persistent sequential scan with grid-wide barrier
  esn_scan<<<dim3(SCAN_NB), dim3(256), 0, stream>>>(W_res, states, ctr);

  // 3) readout (1024 tiles, 4 waves/block)
  esn_readout_wmma<<<dim3(256), dim3(128), 0, stream>>>(inputs, W_out, states, outputs);
}